// Loss_8615704396494
// MI455X (gfx1250) — compile-verified
//
#include <hip/hip_runtime.h>
#include <hip/hip_bf16.h>

typedef __attribute__((ext_vector_type(2))) float v2f;
typedef __attribute__((ext_vector_type(8))) float v8f;

#define N_JOINTS 50
#define ROW_ELEMS 150
#define EPSB 1e-8f

// Cross-lane reduction on the matrix pipe: B = all-ones => D[m][n] = sum_k A[m][k],
// independent of B's VGPR layout. A VGPR0 = per-lane partial, VGPR1 = 0:
//   A[m][0] = x[lane m], A[m][2] = x[lane m+16]  =>  D[m][n] = x[m] + x[m+16].
// Lane l<16 holds D[0..7][n] in its 8 D VGPRs; lane l>=16 holds D[8..15][n].
// Sum of lane 0's D VGPRs + sum of lane 16's D VGPRs = full wave32 sum.
__device__ __forceinline__ float wave_halfsum_wmma(float x) {
    v2f a; a[0] = x;    a[1] = 0.0f;
    v2f b; b[0] = 1.0f; b[1] = 1.0f;
    v8f c = {};
    v8f d = __builtin_amdgcn_wmma_f32_16x16x4_f32(
        /*neg_a=*/false, a, /*neg_b=*/false, b,
        /*c_mod=*/(short)0, c, /*reuse_a=*/false, /*reuse_b=*/false);
    return ((d[0] + d[1]) + (d[2] + d[3])) + ((d[4] + d[5]) + (d[6] + d[7]));
}

__global__ void __launch_bounds__(256)
bone_loss_kernel(const float* __restrict__ preds,
                 const float* __restrict__ targets,
                 float* __restrict__ accum,   // accum[0]=sum|.|, accum[1]=sum(.^2)
                 unsigned n_bones) {
    __shared__ float s_abs;
    __shared__ float s_sq;
    if (threadIdx.x == 0) { s_abs = 0.0f; s_sq = 0.0f; }
    __syncthreads();

    unsigned tid = blockIdx.x * 256u + threadIdx.x;

    float abs_acc = 0.0f;
    float sq_acc  = 0.0f;

    if (tid < n_bones) {
        unsigned row  = tid / N_JOINTS;          // magic-multiply, no real div
        unsigned bone = tid - row * N_JOINTS;
        unsigned base = row * ROW_ELEMS;
        unsigned so   = base + 3u * bone;                                   // SRC joint
        unsigned dofs = base + ((bone == N_JOINTS - 1u) ? 0u : 3u * (bone + 1u)); // DST joint

        // SRC joint loads (coalesced b96-class across the wave)
        float ts0 = targets[so + 0], ts1 = targets[so + 1], ts2 = targets[so + 2];
        float ps0 = preds  [so + 0], ps1 = preds  [so + 1], ps2 = preds  [so + 2];
        // DST joint loads (neighbor lane's SRC line -> L0/L2 hit)
        float td0 = targets[dofs + 0], td1 = targets[dofs + 1], td2 = targets[dofs + 2];
        float pd0 = preds  [dofs + 0], pd1 = preds  [dofs + 1], pd2 = preds  [dofs + 2];

        // loss_mask = (targets != 0)
        float ms0 = (ts0 != 0.0f) ? 1.0f : 0.0f;
        float ms1 = (ts1 != 0.0f) ? 1.0f : 0.0f;
        float ms2 = (ts2 != 0.0f) ? 1.0f : 0.0f;
        float md0 = (td0 != 0.0f) ? 1.0f : 0.0f;
        float md1 = (td1 != 0.0f) ? 1.0f : 0.0f;
        float md2 = (td2 != 0.0f) ? 1.0f : 0.0f;

        // masked values
        float pms0 = ps0 * ms0, pms1 = ps1 * ms1, pms2 = ps2 * ms2;
        float tms0 = ts0 * ms0, tms1 = ts1 * ms1, tms2 = ts2 * ms2;
        float pmd0 = pd0 * md0, pmd1 = pd1 * md1, pmd2 = pd2 * md2;
        float tmd0 = td0 * md0, tmd1 = td1 * md1, tmd2 = td2 * md2;

        // L1 term: each element is SRC of exactly one bone -> counted once
        abs_acc = fabsf(pms0 - tms0) + fabsf(pms1 - tms1) + fabsf(pms2 - tms2);

        // bone vectors from masked joints
        float dpx = pms0 - pmd0, dpy = pms1 - pmd1, dpz = pms2 - pmd2;
        float dtx = tms0 - tmd0, dty = tms1 - tmd1, dtz = tms2 - tmd2;
        float lp = sqrtf(dpx * dpx + dpy * dpy + dpz * dpz);
        float lt = sqrtf(dtx * dtx + dty * dty + dtz * dtz);
        float rp = 1.0f / (lp + EPSB);
        float rt = 1.0f / (lt + EPSB);

        // (pd - td) masked by SRC-element mask, squared
        float e0 = ms0 * (dpx * rp - dtx * rt);
        float e1 = ms1 * (dpy * rp - dty * rt);
        float e2 = ms2 * (dpz * rp - dtz * rt);
        sq_acc = e0 * e0 + e1 * e1 + e2 * e2;
    }
    // guard closed: wave re-converges, EXEC all-1s for WMMA

    float habs = wave_halfsum_wmma(abs_acc);
    float hsq  = wave_halfsum_wmma(sq_acc);

    unsigned lane = threadIdx.x & 31u;
    if (lane == 0u || lane == 16u) {   // the two half-wave totals
        atomicAdd(&s_abs, habs);       // ds_add_f32
        atomicAdd(&s_sq,  hsq);
    }
    __syncthreads();
    if (threadIdx.x == 0) {
        atomicAdd(&accum[0], s_abs);   // global_atomic_add_f32
        atomicAdd(&accum[1], s_sq);
    }
}

__global__ void init_accum_kernel(float* accum) {
    accum[0] = 0.0f;
    accum[1] = 0.0f;
}

__global__ void finalize_kernel(const float* __restrict__ accum,
                                float* __restrict__ out, float inv_n) {
    out[0] = accum[0] * inv_n + 0.1f * (accum[1] * inv_n);
}

extern "C" void kernel_launch(void* const* d_in, const int* in_sizes, int n_in,
                              void* d_out, int out_size, void* d_ws, size_t ws_size,
                              hipStream_t stream) {
    const float* preds   = (const float*)d_in[0];
    const float* targets = (const float*)d_in[1];
    float* out   = (float*)d_out;
    float* accum = (float*)d_ws;

    const unsigned total   = (unsigned)in_sizes[0];          // B*T*150
    const unsigned rows    = total / ROW_ELEMS;              // B*T
    const unsigned n_bones = rows * N_JOINTS;
    const unsigned blocks  = (n_bones + 255u) / 256u;
    const float    inv_n   = 1.0f / (float)total;            // same denominator for both means

    init_accum_kernel<<<1, 1, 0, stream>>>(accum);
    bone_loss_kernel<<<blocks, 256, 0, stream>>>(preds, targets, accum, n_bones);
    finalize_kernel<<<1, 1, 0, stream>>>(accum, out, inv_n);
}